// MLE_41266045780111
// MI455X (gfx1250) — compile-verified
//
#include <hip/hip_runtime.h>

typedef float v2f __attribute__((ext_vector_type(2)));
typedef float v4f __attribute__((ext_vector_type(4)));
typedef float v8f __attribute__((ext_vector_type(8)));

#define B_TOTAL   8388608      // 2^23 elements
#define BLOCKS    2048
#define THREADS   256
#define ITERS     8            // (B/2) pairs / (BLOCKS*THREADS)

#define EPS           1e-5f
#define HALF_LOG_2PI  0.91893853320467274178f
#define INV_SQRT2     0.70710678118654752440f

__device__ __forceinline__ float loss_elem(float mu, float s, float tte, float alive) {
    // sigma = exp(s); z = (log(tte)-mu)/sigma
    float sigma = __expf(s);
    float logt  = __logf(tte);
    float z     = (logt - mu) / sigma;
    float cdf   = 0.5f * (1.0f + erff(z * INV_SQRT2));
    float alive_loss = -__logf(1.0f - cdf + EPS);
    float logx  = __logf(tte + EPS);
    float q     = (logx - mu) / sigma;
    float log_prob  = -logx - s - HALF_LOG_2PI - 0.5f * q * q;
    float dead_loss = -log_prob;
    return alive * alive_loss + (1.0f - alive) * dead_loss;
}

__global__ __launch_bounds__(THREADS)
void mle_partial_kernel(const v4f* __restrict__ pred,   // (B,2) floats as B/2 float4
                        const v4f* __restrict__ tgt,    // (B,2) floats as B/2 float4
                        float* __restrict__ ws)         // BLOCKS partial sums
{
    const int tid    = blockIdx.x * THREADS + threadIdx.x;
    const int stride = BLOCKS * THREADS;

    // Two accumulators per lane -> fills the 16x4 f32 A-matrix (2 VGPRs x 32 lanes).
    float acc0 = 0.0f, acc1 = 0.0f;
#pragma unroll
    for (int k = 0; k < ITERS; ++k) {
        int p = tid + k * stride;                       // pair index, fully coalesced
        v4f pp = __builtin_nontemporal_load(&pred[p]);  // mu0,s0,mu1,s1  (global_load_b128, NT)
        v4f tt = __builtin_nontemporal_load(&tgt[p]);   // tte0,a0,tte1,a1
        acc0 += loss_elem(pp.x, pp.y, tt.x, tt.y);
        acc1 += loss_elem(pp.z, pp.w, tt.z, tt.w);
    }

    // Wave32 reduction via V_WMMA_F32_16X16X4_F32 with a ones B-matrix:
    // D[m][n] = sum_k A[m][k]; summing the 8 D VGPRs per lane gives the sum of
    // rows 0-7 (lanes 0-15) / rows 8-15 (lanes 16-31); one xor-16 shuffle finishes.
    v2f a; a.x = acc0; a.y = acc1;
    v2f b; b.x = 1.0f; b.y = 1.0f;
    v8f c = {};
    c = __builtin_amdgcn_wmma_f32_16x16x4_f32(
        /*neg_a=*/false, a, /*neg_b=*/false, b,
        /*c_mod=*/(short)0, c, /*reuse_a=*/false, /*reuse_b=*/false);
    float part = c[0] + c[1] + c[2] + c[3] + c[4] + c[5] + c[6] + c[7];
    part += __shfl_xor(part, 16, 32);   // combine low/high half-wave partial sums

    __shared__ float smem[THREADS / 32];
    const int lane = threadIdx.x & 31;
    const int wave = threadIdx.x >> 5;
    if (lane == 0) smem[wave] = part;
    __syncthreads();
    if (threadIdx.x == 0) {
        float bs = 0.0f;
#pragma unroll
        for (int w = 0; w < THREADS / 32; ++w) bs += smem[w];
        ws[blockIdx.x] = bs;
    }
}

__global__ __launch_bounds__(THREADS)
void mle_final_kernel(const float* __restrict__ ws, float* __restrict__ out)
{
    float v = 0.0f;
#pragma unroll
    for (int k = 0; k < BLOCKS / THREADS; ++k)
        v += ws[threadIdx.x + k * THREADS];
#pragma unroll
    for (int off = 16; off > 0; off >>= 1)
        v += __shfl_xor(v, off, 32);

    __shared__ float smem[THREADS / 32];
    const int lane = threadIdx.x & 31;
    const int wave = threadIdx.x >> 5;
    if (lane == 0) smem[wave] = v;
    __syncthreads();
    if (threadIdx.x == 0) {
        float total = 0.0f;
#pragma unroll
        for (int w = 0; w < THREADS / 32; ++w) total += smem[w];
        out[0] = total * (1.0f / (float)B_TOTAL);
    }
}

extern "C" void kernel_launch(void* const* d_in, const int* in_sizes, int n_in,
                              void* d_out, int out_size, void* d_ws, size_t ws_size,
                              hipStream_t stream) {
    const v4f* pred = (const v4f*)d_in[0];   // pred_params: (B,2) float32
    const v4f* tgt  = (const v4f*)d_in[1];   // tgts:        (B,2) float32
    float* out = (float*)d_out;
    float* ws  = (float*)d_ws;               // needs BLOCKS*4 = 8 KiB scratch

    mle_partial_kernel<<<BLOCKS, THREADS, 0, stream>>>(pred, tgt, ws);
    mle_final_kernel<<<1, THREADS, 0, stream>>>(ws, out);
}